// StructureTensor3D_79860621902195
// MI455X (gfx1250) — compile-verified
//
#include <hip/hip_runtime.h>
#include <math.h>

// Problem dims (fixed by reference): x (2,4,64,128,128) f32
#define NB 2
#define CH 4
#define DD 64
#define HH 128
#define WW 128

// Output tile per workgroup
#define TZ 4
#define TY 16
#define TX 16

// LDS layout (floats). Region A [0,12800) is phase-aliased.
//  Phase A/B: s_in[4][8][20][20]  = 12800 fl
//  Phase C  : s_u[8][20][18] @0 (2880), s_v @2880 (2880)
//  Phase D  : s_g[3][6][18][18] @5760 (5832) -> ends 11592
//  s_m[8][20][20] @12800 (3200) lives through phase C.
#define SMEM_FLOATS 16000
#define SIN_ELEMS   12800
#define SM_OFF      12800
#define V_OFF       2880
#define G_OFF       5760

typedef float v2f __attribute__((ext_vector_type(2)));
typedef float v8f __attribute__((ext_vector_type(8)));

#if defined(__gfx1250__) && __has_builtin(__builtin_amdgcn_global_load_async_to_lds_b32)
#define HAVE_ASYNC 1
#else
#define HAVE_ASYNC 0
#endif

#if defined(__gfx1250__) && __has_builtin(__builtin_amdgcn_wmma_f32_16x16x4_f32)
#define HAVE_WMMA 1
#else
#define HAVE_WMMA 0
#endif

__device__ __forceinline__ float finite_or_zero(float v) {
  // true for finite v; inf*0 = nan, nan*0 = nan
  return (v * 0.0f == 0.0f) ? v : 0.0f;
}

__global__ __launch_bounds__(256) void st3d_fused(const float* __restrict__ x,
                                                  float* __restrict__ out_p,
                                                  float* __restrict__ out_c) {
  __shared__ __align__(16) float smem[SMEM_FLOATS];
  float* s_in = smem;            // phase A/B
  float* s_m  = smem + SM_OFF;   // channel mean 8x20x20
  float* s_u  = smem;            // x-smoothed 8x20x18  (aliases s_in)
  float* s_v  = smem + V_OFF;    // x-deriv    8x20x18
  float* s_g  = smem + G_OFF;    // gradients 3x(6x18x18)

  const int tid = threadIdx.x;
  const int w0 = blockIdx.x * TX;
  const int h0 = blockIdx.y * TY;
  const int bz = blockIdx.z;
  const int n  = bz >> 4;           // DD/TZ = 16 depth tiles
  const int d0 = (bz & 15) * TZ;

  // ---------------- Phase A: stage 4-channel halo tile (8x20x20) into LDS ----
  for (int e = tid; e < SIN_ELEMS; e += 256) {
    int ch = e / 3200;
    int r  = e - ch * 3200;
    int z  = r / 400;
    int r2 = r - z * 400;
    int yy = r2 / 20;
    int xl = r2 - yy * 20;
    int zg = d0 - 2 + z;
    int yg = h0 - 2 + yy;
    int xg = w0 - 2 + xl;
    bool ok = (zg >= 0) & (zg < DD) & (yg >= 0) & (yg < HH) & (xg >= 0) & (xg < WW);
    if (ok) {
      const float* gp = x + ((((size_t)n * CH + ch) * DD + zg) * HH + yg) * WW + xg;
#if HAVE_ASYNC
      __builtin_amdgcn_global_load_async_to_lds_b32(
          (__attribute__((address_space(1))) int*)gp,
          (__attribute__((address_space(3))) int*)(s_in + e),
          /*offset=*/0, /*cpol=*/0);
#else
      s_in[e] = *gp;
#endif
    } else {
      s_in[e] = 0.0f;   // zero padding (DS store; disjoint address from async writes)
    }
  }
#if HAVE_ASYNC
#if __has_builtin(__builtin_amdgcn_s_wait_asynccnt)
  __builtin_amdgcn_s_wait_asynccnt(0);
#else
  asm volatile("s_wait_asynccnt 0x0" ::: "memory");
#endif
#endif
  __syncthreads();

  // ---------------- Phase B: channel mean -> s_m (8x20x20) -------------------
  for (int e = tid; e < 3200; e += 256) {
    s_m[e] = 0.25f * (s_in[e] + s_in[e + 3200] + s_in[e + 6400] + s_in[e + 9600]);
  }
  __syncthreads();

  // ---------------- Phase C: x-axis separable pass via WMMA ------------------
  // u = [1,2,1]_x * m ; v = [-1,0,1]_x * m, on grid z:[0,8) y:[0,20) c:[0,18)
  // Out(16x16) = In(16x20) x W(20x16), W banded: W(Ki,c) = coef[Ki-c], Ki-c in {0,1,2}
#if HAVE_WMMA
  {
    const int wave = tid >> 5;
    const int lane = tid & 31;
    const int half = lane >> 4;   // 0 or 1
    const int l16  = lane & 15;
    // tasks: f in {0,1} (smooth/deriv) x mt in [0,10) x nt in {0,1} = 40
    for (int task = wave; task < 40; task += 8) {
      const int f  = task / 20;
      const int r  = task - f * 20;
      const int mt = r >> 1;
      const int nt = r & 1;
      const int c  = nt * 16 + l16;     // output column (valid if < 18)
      v8f acc = {0.f, 0.f, 0.f, 0.f, 0.f, 0.f, 0.f, 0.f};
      for (int k = 0; k < 5; ++k) {
        // A fragment: lane holds (M = l16, K = kb, kb+1)
        const int Ra = mt * 16 + l16;       // flattened (z,y) row, < 160
        const int za = Ra / 20;
        const int ya = Ra - za * 20;
        const int kb = k * 4 + half * 2;    // <= 19
        v2f A;
        A.x = s_m[za * 400 + ya * 20 + kb];
        A.y = s_m[za * 400 + ya * 20 + kb + 1];
        // B fragment: lane holds (K = kb, kb+1 ; N = l16)
        v2f B;
        {
          float w0_ = 0.0f, w1_ = 0.0f;
          int t0 = kb - c, t1 = kb + 1 - c;
          if (c < 18) {
            if (t0 >= 0 && t0 <= 2)
              w0_ = (f == 0) ? ((t0 == 1) ? 2.0f : 1.0f)
                             : ((t0 == 0) ? -1.0f : ((t0 == 2) ? 1.0f : 0.0f));
            if (t1 >= 0 && t1 <= 2)
              w1_ = (f == 0) ? ((t1 == 1) ? 2.0f : 1.0f)
                             : ((t1 == 0) ? -1.0f : ((t1 == 2) ? 1.0f : 0.0f));
          }
          B.x = w0_; B.y = w1_;
        }
        acc = __builtin_amdgcn_wmma_f32_16x16x4_f32(false, A, false, B,
                                                    (short)0, acc, false, false);
      }
      float* dst = (f == 0) ? s_u : s_v;
      if (c < 18) {
        #pragma unroll
        for (int e2 = 0; e2 < 8; ++e2) {
          const int M  = e2 + half * 8;
          const int Rr = mt * 16 + M;
          const int zr = Rr / 20;
          const int yr = Rr - zr * 20;
          dst[zr * 360 + yr * 18 + c] = acc[e2];
        }
      }
    }
  }
#else
  for (int e = tid; e < 2880 * 2; e += 256) {
    int f = e / 2880;
    int r = e - f * 2880;
    int z = r / 360;
    int r2 = r - z * 360;
    int yy = r2 / 18;
    int c  = r2 - yy * 18;
    const int base = z * 400 + yy * 20 + c;
    float m0 = s_m[base], m1 = s_m[base + 1], m2 = s_m[base + 2];
    ((f == 0) ? s_u : s_v)[z * 360 + yy * 18 + c] =
        (f == 0) ? (m0 + 2.0f * m1 + m2) : (m2 - m0);
  }
#endif
  __syncthreads();

  // ---------------- Phase D: (y,z) passes -> gradients (6x18x18) -------------
  for (int e = tid; e < 1944; e += 256) {
    int a = e / 324;
    int r = e - a * 324;
    int b = r / 18;
    int c = r - b * 18;
    float gx = 0.f, gy = 0.f, gz = 0.f;
    #pragma unroll
    for (int tz = 0; tz < 3; ++tz) {
      const float sz = (tz == 1) ? 2.0f : 1.0f;
      const float dz = (tz == 0) ? -1.0f : ((tz == 2) ? 1.0f : 0.0f);
      #pragma unroll
      for (int ty = 0; ty < 3; ++ty) {
        const float sy = (ty == 1) ? 2.0f : 1.0f;
        const float dy = (ty == 0) ? -1.0f : ((ty == 2) ? 1.0f : 0.0f);
        const int idx = (a + tz) * 360 + (b + ty) * 18 + c;
        const float uu = s_u[idx];
        const float vv = s_v[idx];
        gx += dz * sy * uu;   // d_z s_y s_x
        gy += sz * dy * uu;   // s_z d_y s_x
        gz += sz * sy * vv;   // s_z s_y d_x
      }
    }
    s_g[e]        = gx * 0.125f;
    s_g[e + 1944] = gy * 0.125f;
    s_g[e + 3888] = gz * 0.125f;
  }
  __syncthreads();

  // ---------------- Phase E: pool products, 3x3 eigen, write -----------------
  for (int e = tid; e < TZ * TY * TX; e += 256) {
    int oz = e >> 8;
    int r  = e & 255;
    int oy = r >> 4;
    int ox = r & 15;
    float xx = 0, xy = 0, xz = 0, yy = 0, yz = 0, zz = 0;
    #pragma unroll
    for (int tz = 0; tz < 3; ++tz)
      #pragma unroll
      for (int ty = 0; ty < 3; ++ty)
        #pragma unroll
        for (int tx = 0; tx < 3; ++tx) {
          const int idx = (oz + tz) * 324 + (oy + ty) * 18 + (ox + tx);
          const float a_ = s_g[idx];
          const float b_ = s_g[idx + 1944];
          const float c_ = s_g[idx + 3888];
          xx += a_ * a_; xy += a_ * b_; xz += a_ * c_;
          yy += b_ * b_; yz += b_ * c_; zz += c_ * c_;
        }
    const float inv27 = 1.0f / 27.0f;
    xx *= inv27; xy *= inv27; xz *= inv27;
    yy *= inv27; yz *= inv27; zz *= inv27;

    const float trace = xx + yy + zz;
    const float epsr  = trace * (1.0f / 3.0f) * 1e-3f + 1e-6f;
    float axx = finite_or_zero(xx + epsr);
    float ayy = finite_or_zero(yy + epsr);
    float azz = finite_or_zero(zz + epsr);
    xy = finite_or_zero(xy); xz = finite_or_zero(xz); yz = finite_or_zero(yz);
    const bool low = (trace <= 1e-6f);

    const float q  = (axx + ayy + azz) * (1.0f / 3.0f);
    const float p1 = xy * xy + xz * xz + yz * yz;
    const float dxx = axx - q, dyy = ayy - q, dzz = azz - q;
    const float p2 = dxx * dxx + dyy * dyy + dzz * dzz + 2.0f * p1;

    float e1, e2v, e3;
    float vx = 0.f, vy = 0.f, vz = 0.f;
    if (p2 > 1e-30f) {
      const float p  = sqrtf(p2 * (1.0f / 6.0f));
      const float ip = 1.0f / p;
      const float bxx = dxx * ip, byy = dyy * ip, bzz = dzz * ip;
      const float bxy = xy * ip,  bxz = xz * ip,  byz = yz * ip;
      float detB = bxx * (byy * bzz - byz * byz)
                 - bxy * (bxy * bzz - byz * bxz)
                 + bxz * (bxy * byz - byy * bxz);
      float rr = fminf(1.0f, fmaxf(-1.0f, 0.5f * detB));
      const float phi = acosf(rr) * (1.0f / 3.0f);
      e1  = q + 2.0f * p * cosf(phi);
      e3  = q + 2.0f * p * cosf(phi + 2.0943951023931953f);  // +2*pi/3
      e2v = 3.0f * q - e1 - e3;
      // eigenvector of smallest eigenvalue e3: largest cross of rows of (A - e3 I)
      const float m00 = axx - e3, m11 = ayy - e3, m22 = azz - e3;
      float c0x = xy * yz - m11 * xz, c0y = xz * xy - m00 * yz, c0z = m00 * m11 - xy * xy;
      float c1x = xy * m22 - xz * yz, c1y = xz * xz - m00 * m22, c1z = m00 * yz - xy * xz;
      float c2x = m11 * m22 - yz * yz, c2y = yz * xz - xy * m22, c2z = xy * yz - m11 * xz;
      float n0 = c0x * c0x + c0y * c0y + c0z * c0z;
      float n1 = c1x * c1x + c1y * c1y + c1z * c1z;
      float n2 = c2x * c2x + c2y * c2y + c2z * c2z;
      float bx = c0x, by = c0y, bzv = c0z, bn = n0;
      if (n1 > bn) { bx = c1x; by = c1y; bzv = c1z; bn = n1; }
      if (n2 > bn) { bx = c2x; by = c2y; bzv = c2z; bn = n2; }
      const float nrm = sqrtf(bn);
      if (nrm > 1e-8f) {
        const float inr = 1.0f / nrm;
        vx = bx * inr; vy = by * inr; vz = bzv * inr;
      }
    } else {
      e1 = e2v = e3 = q;
      vx = 1.0f; vy = 0.0f; vz = 0.0f;   // eigh(c*I) -> identity columns
    }
    if (low) { vx = vy = vz = 0.0f; }
    vx = finite_or_zero(vx); vy = finite_or_zero(vy); vz = finite_or_zero(vz);
    const float l1 = low ? 0.f : finite_or_zero(e1);
    const float l2 = low ? 0.f : finite_or_zero(e2v);
    const float l3 = low ? 0.f : finite_or_zero(e3);
    const float coher = (l2 - l3) / (l1 + l2 + l3 + 1e-6f);

    const size_t vox = (((size_t)n * DD + (d0 + oz)) * HH + (h0 + oy)) * WW + (w0 + ox);
    out_p[vox * 3 + 0] = vx;
    out_p[vox * 3 + 1] = vy;
    out_p[vox * 3 + 2] = vz;
    out_c[vox] = coher;
  }
}

extern "C" void kernel_launch(void* const* d_in, const int* in_sizes, int n_in,
                              void* d_out, int out_size, void* d_ws, size_t ws_size,
                              hipStream_t stream) {
  (void)in_sizes; (void)n_in; (void)out_size; (void)d_ws; (void)ws_size;
  const float* x = (const float*)d_in[0];
  float* out_p = (float*)d_out;                              // (2,64,128,128,3)
  float* out_c = out_p + (size_t)NB * DD * HH * WW * 3;      // (2,64,128,128)
  dim3 grid(WW / TX, HH / TY, NB * (DD / TZ));               // 8 x 8 x 32
  st3d_fused<<<grid, dim3(256), 0, stream>>>(x, out_p, out_c);
}